// FusedGWLoss_14937896256169
// MI455X (gfx1250) — compile-verified
//
#include <hip/hip_runtime.h>
#include <math.h>
#include <stdint.h>

typedef __attribute__((ext_vector_type(2))) float v2f;
typedef __attribute__((ext_vector_type(8))) float v8f;

#define N1C 2048
#define N2C 1536
#define DC  128
#define EPSC 0.01f
#define IEPS 100.0f
#define GWC  20.0f

enum { POST_NONE = 0, POST_EXPNEG = 1, POST_EXPNEG_ADJ = 2, POST_Q = 3, POST_SUBR1 = 4 };

// ---------------------------------------------------------------------------
// CDNA5 async global->LDS copy (ASYNCcnt-tracked, bypasses VGPRs)
// ---------------------------------------------------------------------------
__device__ __forceinline__ void async_copy_b128(uint32_t lds_off, const void* gptr) {
  uint64_t ga = (uint64_t)(uintptr_t)gptr;
  asm volatile("global_load_async_to_lds_b128 %0, %1, off"
               :: "v"(lds_off), "v"(ga)
               : "memory");
}

template<int N>
__device__ __forceinline__ void wait_asynccnt() {
  asm volatile("s_wait_asynccnt %0" :: "i"(N) : "memory");
}

// ---------------------------------------------------------------------------
// Block reduction helpers (deterministic tree reductions through LDS)
// ---------------------------------------------------------------------------
__device__ __forceinline__ float blk_sum256(float v, float* red) {
  int t = threadIdx.x;
  red[t] = v; __syncthreads();
  for (int s = 128; s > 0; s >>= 1) {
    if (t < s) red[t] += red[t + s];
    __syncthreads();
  }
  float r = red[0]; __syncthreads();
  return r;
}

__device__ __forceinline__ float blk_max256(float v, float* red) {
  int t = threadIdx.x;
  red[t] = v; __syncthreads();
  for (int s = 128; s > 0; s >>= 1) {
    if (t < s) red[t] = fmaxf(red[t], red[t + s]);
    __syncthreads();
  }
  float r = red[0]; __syncthreads();
  return r;
}

// ---------------------------------------------------------------------------
// WMMA f32 GEMM: C = A[M,K] @ B (NN) or A @ B^T (NT), fused epilogues.
// 128x128 block tile, 8 waves (wave32), each wave 32x64 via 2x4 16x16 tiles.
// V_WMMA_F32_16X16X4_F32 + double-buffered async global->LDS staging.
// ---------------------------------------------------------------------------
template<int POST, bool TRANSB>
__global__ __launch_bounds__(256)
void gemm_wmma_f32(const float* __restrict__ A, const float* __restrict__ B,
                   float* __restrict__ C, int K, int lda, int ldb, int ldc,
                   const float* __restrict__ lamPtr,  // SUBR1: lambda scalar
                   const float* __restrict__ r1vec,   // SUBR1: rowsum(C1)
                   const float* __restrict__ aux,     // ADJ: adj | Q: inter_c
                   const float* __restrict__ uvec,    // Q: u[row]
                   const float* __restrict__ vvec)    // Q: v[col]
{
  __shared__ float As[2][128 * 20];                       // [m][k] (+4 pad)
  __shared__ float Bs[2][TRANSB ? 128 * 20 : 16 * 132];   // NT:[n][k]  NN:[k][n]

  const int tid  = threadIdx.x;
  const int lane = tid & 31;
  const int wid  = tid >> 5;       // 0..7
  const int wm   = wid >> 1;       // 0..3  (wave row)
  const int wn   = wid & 1;        // 0..1  (wave col)
  const int half = lane >> 4;      // K/M lane-group select per ISA layout
  const int l    = lane & 15;
  const int m0 = blockIdx.y * 128;
  const int n0 = blockIdx.x * 128;

  v8f acc[2][4];
#pragma unroll
  for (int mt = 0; mt < 2; ++mt)
#pragma unroll
    for (int nt = 0; nt < 4; ++nt)
#pragma unroll
      for (int r = 0; r < 8; ++r) acc[mt][nt][r] = 0.0f;

  // stage one 16-wide K slab into LDS buffer `buf` (4 async b128 per thread)
  auto stage = [&](int k0, int buf) {
#pragma unroll
    for (int rep = 0; rep < 2; ++rep) {
      int t = tid + rep * 256;
      int row = t >> 2, c4 = (t & 3) << 2;
      async_copy_b128((uint32_t)(uintptr_t)&As[buf][row * 20 + c4],
                      A + (size_t)(m0 + row) * lda + k0 + c4);
    }
    if (!TRANSB) {
#pragma unroll
      for (int rep = 0; rep < 2; ++rep) {
        int t = tid + rep * 256;
        int krow = t >> 5, c4 = (t & 31) << 2;
        async_copy_b128((uint32_t)(uintptr_t)&Bs[buf][krow * 132 + c4],
                        B + (size_t)(k0 + krow) * ldb + n0 + c4);
      }
    } else {
#pragma unroll
      for (int rep = 0; rep < 2; ++rep) {
        int t = tid + rep * 256;
        int nn = t >> 2, k4 = (t & 3) << 2;
        async_copy_b128((uint32_t)(uintptr_t)&Bs[buf][nn * 20 + k4],
                        B + (size_t)(n0 + nn) * ldb + k0 + k4);
      }
    }
  };

  auto compute = [&](int buf) {
#pragma unroll
    for (int kk = 0; kk < 16; kk += 4) {
      const int ka = kk + 2 * half;
      v2f fa[2], fb[4];
#pragma unroll
      for (int mt = 0; mt < 2; ++mt) {
        int row = wm * 32 + mt * 16 + l;
        fa[mt].x = As[buf][row * 20 + ka];
        fa[mt].y = As[buf][row * 20 + ka + 1];
      }
#pragma unroll
      for (int nt = 0; nt < 4; ++nt) {
        int col = wn * 64 + nt * 16 + l;
        if (!TRANSB) {
          fb[nt].x = Bs[buf][ka * 132 + col];
          fb[nt].y = Bs[buf][(ka + 1) * 132 + col];
        } else {
          fb[nt].x = Bs[buf][col * 20 + ka];
          fb[nt].y = Bs[buf][col * 20 + ka + 1];
        }
      }
#pragma unroll
      for (int mt = 0; mt < 2; ++mt)
#pragma unroll
        for (int nt = 0; nt < 4; ++nt)
          acc[mt][nt] = __builtin_amdgcn_wmma_f32_16x16x4_f32(
              false, fa[mt], false, fb[nt], (short)0, acc[mt][nt], false, false);
    }
  };

  const int nslabs = K >> 4;
  stage(0, 0);
  for (int i = 0; i < nslabs; ++i) {
    const int buf = i & 1;
    if (i + 1 < nslabs) {
      stage((i + 1) << 4, buf ^ 1);   // prefetch next slab into other buffer
      wait_asynccnt<4>();             // oldest 4 (current slab) complete
    } else {
      wait_asynccnt<0>();
    }
    __syncthreads();
    compute(buf);
    __syncthreads();                  // done reading `buf` before it is rewritten
  }

  // ---- epilogue (C/D layout: VGPR r -> M = r + 8*half, N = l) ----
  float lamv = 0.0f;
  if (POST == POST_SUBR1) lamv = lamPtr[0];
#pragma unroll
  for (int mt = 0; mt < 2; ++mt)
#pragma unroll
    for (int nt = 0; nt < 4; ++nt) {
      const int rbase = m0 + wm * 32 + mt * 16 + half * 8;
      const int col   = n0 + wn * 64 + nt * 16 + l;
#pragma unroll
      for (int r = 0; r < 8; ++r) {
        float x = acc[mt][nt][r];
        size_t idx = (size_t)(rbase + r) * ldc + col;
        if (POST == POST_NONE)            C[idx] = x;
        else if (POST == POST_EXPNEG)     C[idx] = __expf(-x);
        else if (POST == POST_EXPNEG_ADJ) C[idx] = __expf(-x) * aux[idx];
        else if (POST == POST_SUBR1)      C[idx] = x - lamv * r1vec[rbase + r];
        else /* POST_Q */                 C[idx] = aux[idx] + GWC * (uvec[rbase + r] + vvec[col] - 2.0f * x);
      }
    }
}

// ---------------------------------------------------------------------------
// Row reduction: out[i] = sum_j X[i,j]^(1|2) * (w?w[j]:1) + bias  (block/row)
// ---------------------------------------------------------------------------
template<bool SQ>
__global__ __launch_bounds__(256)
void rowreduce_kernel(const float* __restrict__ X, const float* __restrict__ w,
                      float* __restrict__ out, int cols, float bias) {
  __shared__ float red[256];
  const float* x = X + (size_t)blockIdx.x * cols;
  float s = 0.0f;
  for (int j = threadIdx.x; j < cols; j += 256) {
    float v = x[j];
    if (SQ) v *= v;
    if (w) v *= w[j];
    s += v;
  }
  float t = blk_sum256(s, red);
  if (threadIdx.x == 0) out[blockIdx.x] = t + bias;
}

// Column reduction: out[j] = sum_i X[i,j]^(1|2) * (w?w[i]:1) + bias
template<bool SQ>
__global__ void colreduce_kernel(const float* __restrict__ X, const float* __restrict__ w,
                                 float* __restrict__ out, int rows, int cols, float bias) {
  int j = blockIdx.x * 256 + threadIdx.x;
  if (j >= cols) return;
  float s = 0.0f;
  for (int i = 0; i < rows; ++i) {
    float v = X[(size_t)i * cols + j];
    if (SQ) v *= v;
    if (w) v *= w[i];
    s += v;
  }
  out[j] = s + bias;
}

// ---------------------------------------------------------------------------
// Sinkhorn soft-min kernels (stable logsumexp), eps = 0.01
// ---------------------------------------------------------------------------
__global__ __launch_bounds__(256)
void lse_row_kernel(const float* __restrict__ Q, const float* __restrict__ g,
                    float* __restrict__ f, int cols, float logab) {
  __shared__ float red[256];
  const float* q = Q + (size_t)blockIdx.x * cols;
  float vals[8];
  int cnt = 0;
  float mx = -3.4e38f;
  for (int j = threadIdx.x; j < cols; j += 256) {
    float x = (g[j] - q[j]) * IEPS;
    vals[cnt++] = x;
    mx = fmaxf(mx, x);
  }
  mx = blk_max256(mx, red);
  float s = 0.0f;
  for (int t = 0; t < cnt; ++t) s += __expf(vals[t] - mx);
  s = blk_sum256(s, red);
  if (threadIdx.x == 0) f[blockIdx.x] = -EPSC * (mx + logf(s)) + logab;
}

__global__ void lse_col_kernel(const float* __restrict__ Q, const float* __restrict__ f,
                               float* __restrict__ g, int rows, int cols, float logb) {
  int j = blockIdx.x * 256 + threadIdx.x;
  if (j >= cols) return;
  float mx = -3.4e38f, s = 0.0f;
  for (int i = 0; i < rows; ++i) {
    float x = (f[i] - Q[(size_t)i * cols + j]) * IEPS;
    if (x > mx) { s = s * __expf(mx - x) + 1.0f; mx = x; }
    else        { s += __expf(x - mx); }
  }
  g[j] = -EPSC * (mx + logf(s)) + logb;
}

__global__ void s_update_kernel(float* __restrict__ S, const float* __restrict__ Q,
                                const float* __restrict__ f, const float* __restrict__ g,
                                int cols, size_t n) {
  size_t idx = (size_t)blockIdx.x * 256 + threadIdx.x;
  if (idx >= n) return;
  int i = (int)(idx / cols), j = (int)(idx % cols);
  float e = __expf((f[i] + g[j] - Q[idx]) * IEPS);
  S[idx] = 0.05f * S[idx] + 0.95f * e;
}

// ---------------------------------------------------------------------------
// Misc elementwise / reduction kernels
// ---------------------------------------------------------------------------
__global__ void fill_kernel(float* __restrict__ p, float v, size_t n) {
  size_t idx = (size_t)blockIdx.x * 256 + threadIdx.x;
  if (idx < n) p[idx] = v;
}

__global__ void sub_scalar_kernel(float* __restrict__ dst, const float* __restrict__ src,
                                  const float* __restrict__ sc, size_t n) {
  size_t idx = (size_t)blockIdx.x * 256 + threadIdx.x;
  if (idx < n) dst[idx] = src[idx] - sc[0];
}

__global__ void copy_kernel(float* __restrict__ dst, const float* __restrict__ src, size_t n) {
  size_t idx = (size_t)blockIdx.x * 256 + threadIdx.x;
  if (idx < n) dst[idx] = src[idx];
}

__global__ __launch_bounds__(256)
void bigdot_stage1(const float* __restrict__ X, const float* __restrict__ Y,
                   float* __restrict__ partial, size_t n) {
  __shared__ float red[256];
  float s = 0.0f;
  for (size_t idx = (size_t)blockIdx.x * 256 + threadIdx.x; idx < n; idx += (size_t)gridDim.x * 256)
    s += Y ? X[idx] * Y[idx] : X[idx];
  float t = blk_sum256(s, red);
  if (threadIdx.x == 0) partial[blockIdx.x] = t;
}

__global__ void reduce_final_kernel(const float* __restrict__ partial, int n, float* __restrict__ out) {
  __shared__ float red[256];
  float s = 0.0f;
  for (int i = threadIdx.x; i < n; i += 256) s += partial[i];
  float t = blk_sum256(s, red);
  if (threadIdx.x == 0) out[0] = t;
}

__global__ __launch_bounds__(256)
void gw_stage1(const float* __restrict__ T2, const float* __restrict__ sh,
               const float* __restrict__ u2, const float* __restrict__ v2,
               float* __restrict__ partial, int cols, size_t n) {
  __shared__ float red[256];
  float s = 0.0f;
  for (size_t idx = (size_t)blockIdx.x * 256 + threadIdx.x; idx < n; idx += (size_t)gridDim.x * 256) {
    int i = (int)(idx / cols), j = (int)(idx % cols);
    s += (u2[i] + v2[j] - 2.0f * T2[idx]) * sh[idx];
  }
  float t = blk_sum256(s, red);
  if (threadIdx.x == 0) partial[blockIdx.x] = t;
}

__global__ void dotv_kernel(const float* __restrict__ x, const float* __restrict__ y,
                            float* __restrict__ out, int n) {
  __shared__ float red[256];
  float s = 0.0f;
  for (int i = threadIdx.x; i < n; i += 256) s += y ? x[i] * y[i] : x[i];
  float t = blk_sum256(s, red);
  if (threadIdx.x == 0) out[0] = t;
}

// sc layout: 0=lam 1=k1 2=ta 3=tb 4=tc 5=td 6=te 7=tf 8=tg 9=wl 10=gl
__global__ void lam_finalize_kernel(float* __restrict__ sc, float lam0) {
  if (threadIdx.x == 0 && blockIdx.x == 0) {
    float k1 = sc[1];
    float k2 = (float)N2C * sc[2] + (float)N1C * sc[3] - 2.0f * sc[4];
    float k3 = (float)N2C * (float)N2C * sc[5] + (float)N1C * (float)N1C * sc[6]
             - 2.0f * sc[7] * sc[8];
    float upd = (k1 + 2.0f * GWC * k2) / (2.0f * GWC * k3);
    sc[0] = 0.05f * upd + 0.95f * lam0;
  }
}

__global__ void loss_finalize_kernel(const float* __restrict__ sc, float* __restrict__ out,
                                     size_t ne) {
  if (threadIdx.x == 0 && blockIdx.x == 0) {
    out[0]      = sc[9] + GWC * sc[10] + 20.0f;
    out[1 + ne] = sc[0];
  }
}

// ---------------------------------------------------------------------------
// Host launch sequence (fixed & deterministic; all on `stream`)
// ---------------------------------------------------------------------------
extern "C" void kernel_launch(void* const* d_in, const int* in_sizes, int n_in,
                              void* d_out, int out_size, void* d_ws, size_t ws_size,
                              hipStream_t stream) {
  (void)in_sizes; (void)n_in; (void)out_size; (void)ws_size;
  const float* out1 = (const float*)d_in[0];
  const float* out2 = (const float*)d_in[1];
  const float* adj1 = (const float*)d_in[2];
  const float* adj2 = (const float*)d_in[3];
  float* out = (float*)d_out;

  const size_t NE = (size_t)N1C * N2C;
  const size_t E1 = (size_t)N1C * N1C;
  const size_t E2 = (size_t)N2C * N2C;

  float* w = (float*)d_ws;
  float* interC = w; w += NE;
  float* C1 = w; w += E1;
  float* C2 = w; w += E2;
  float* S  = w; w += NE;
  float* T1 = w; w += NE;
  float* Q  = w; w += NE;   // also T2 at the end
  float* SH = w; w += NE;
  float* fv = w; w += N1C;
  float* gv = w; w += N2C;
  float* ah = w; w += N1C;
  float* bh = w; w += N2C;
  float* uu = w; w += N1C;
  float* vv = w; w += N2C;
  float* r1sq = w; w += N1C;
  float* r1   = w; w += N1C;
  float* c2sqc= w; w += N2C;
  float* r2   = w; w += N2C;
  float* srow = w; w += N1C;
  float* scol = w; w += N2C;
  float* sr2  = w; w += N1C;
  float* u2v  = w; w += N1C;
  float* v2v  = w; w += N2C;
  float* afv  = w; w += N1C;
  float* bfv  = w; w += N2C;
  float* part = w; w += 1024;
  float* sc   = w; w += 32;

  const dim3 blk(256);
  auto gg = [](int M, int N) { return dim3((unsigned)(N / 128), (unsigned)(M / 128)); };
  auto g1 = [](size_t n) { return dim3((unsigned)((n + 255) / 256)); };

  const float lam0  = 1.0f / ((float)N1C * (float)N2C);
  const float log_a = EPSC * logf(1.0f / (float)N1C);
  const float log_b = EPSC * logf(1.0f / (float)N2C);

  // ---- setup: cost matrices via WMMA (NT gemms over d=128) ----
  gemm_wmma_f32<POST_EXPNEG, true><<<gg(N1C, N2C), blk, 0, stream>>>(
      out1, out2, interC, DC, DC, DC, N2C, nullptr, nullptr, nullptr, nullptr, nullptr);
  gemm_wmma_f32<POST_EXPNEG_ADJ, true><<<gg(N1C, N1C), blk, 0, stream>>>(
      out1, out1, C1, DC, DC, DC, N1C, nullptr, nullptr, adj1, nullptr, nullptr);
  gemm_wmma_f32<POST_EXPNEG_ADJ, true><<<gg(N2C, N2C), blk, 0, stream>>>(
      out2, out2, C2, DC, DC, DC, N2C, nullptr, nullptr, adj2, nullptr, nullptr);

  fill_kernel<<<g1(NE), blk, 0, stream>>>(S, lam0, NE);       // s0 = 1/(n1*n2)
  fill_kernel<<<g1(N1C), blk, 0, stream>>>(fv, 1.0f / N1C, N1C);
  fill_kernel<<<g1(N2C), blk, 0, stream>>>(gv, 1.0f / N2C, N2C);
  fill_kernel<<<1, blk, 0, stream>>>(sc, lam0, 1);            // lam (device scalar)

  // r1 = rowsum(C1), loop-invariant (used for the algebraic s-lam fold + later)
  rowreduce_kernel<false><<<N1C, blk, 0, stream>>>(C1, nullptr, r1, N1C, 0.0f);

  // ---- outer Sinkhorn loop (10 iterations) ----
  for (int it = 0; it < 10; ++it) {
    rowreduce_kernel<false><<<N1C, blk, 0, stream>>>(S, nullptr, ah, N2C, -(float)N2C * lam0);
    colreduce_kernel<false><<<g1(N2C), blk, 0, stream>>>(S, nullptr, bh, N1C, N2C, -(float)N1C * lam0);
    // T1 = C1 @ (s - lam) = C1 @ s - lam * r1[i]   (rank-1 epilogue fold)
    gemm_wmma_f32<POST_SUBR1, false><<<gg(N1C, N2C), blk, 0, stream>>>(
        C1, S, T1, N1C, N1C, N2C, N2C, sc, r1, nullptr, nullptr, nullptr);
    rowreduce_kernel<true><<<N1C, blk, 0, stream>>>(C1, ah, uu, N1C, 0.0f);          // u = C1^2 @ a_hat
    colreduce_kernel<true><<<g1(N2C), blk, 0, stream>>>(C2, bh, vv, N2C, N2C, 0.0f); // v = b_hat @ C2^2
    // Q = inter + GW*(u + v - 2 * T1 @ C2^T), fused epilogue
    gemm_wmma_f32<POST_Q, true><<<gg(N1C, N2C), blk, 0, stream>>>(
        T1, C2, Q, N2C, N2C, N2C, N2C, nullptr, nullptr, interC, uu, vv);
    for (int j = 0; j < 5; ++j) {
      lse_row_kernel<<<N1C, blk, 0, stream>>>(Q, gv, fv, N2C, log_a);
      lse_col_kernel<<<g1(N2C), blk, 0, stream>>>(Q, fv, gv, N1C, N2C, log_b);
    }
    s_update_kernel<<<g1(NE), blk, 0, stream>>>(S, Q, fv, gv, N2C, NE);
  }

  // ---- lambda update (rank-1 structure -> vector reductions only) ----
  rowreduce_kernel<true ><<<N1C, blk, 0, stream>>>(C1, nullptr, r1sq, N1C, 0.0f);
  colreduce_kernel<true ><<<g1(N2C), blk, 0, stream>>>(C2, nullptr, c2sqc, N2C, N2C, 0.0f);
  rowreduce_kernel<false><<<N2C, blk, 0, stream>>>(C2, nullptr, r2,   N2C, 0.0f);
  rowreduce_kernel<false><<<N1C, blk, 0, stream>>>(S, nullptr, srow, N2C, 0.0f);
  colreduce_kernel<false><<<g1(N2C), blk, 0, stream>>>(S, nullptr, scol, N1C, N2C, 0.0f);
  rowreduce_kernel<false><<<N1C, blk, 0, stream>>>(S, r2, sr2, N2C, 0.0f);
  bigdot_stage1<<<256, blk, 0, stream>>>(interC, nullptr, part, NE);
  reduce_final_kernel<<<1, blk, 0, stream>>>(part, 256, sc + 1);        // k1
  dotv_kernel<<<1, blk, 0, stream>>>(r1sq, srow, sc + 2, N1C);
  dotv_kernel<<<1, blk, 0, stream>>>(c2sqc, scol, sc + 3, N2C);
  dotv_kernel<<<1, blk, 0, stream>>>(r1, sr2, sc + 4, N1C);
  dotv_kernel<<<1, blk, 0, stream>>>(r1sq, nullptr, sc + 5, N1C);
  dotv_kernel<<<1, blk, 0, stream>>>(c2sqc, nullptr, sc + 6, N2C);
  dotv_kernel<<<1, blk, 0, stream>>>(r1, nullptr, sc + 7, N1C);
  dotv_kernel<<<1, blk, 0, stream>>>(r2, nullptr, sc + 8, N2C);
  lam_finalize_kernel<<<1, 32, 0, stream>>>(sc, lam0);                  // sc[0] = lam_new

  // ---- final loss ----
  sub_scalar_kernel<<<g1(NE), blk, 0, stream>>>(SH, S, sc, NE);         // s_hat = s - lam
  bigdot_stage1<<<256, blk, 0, stream>>>(interC, SH, part, NE);
  reduce_final_kernel<<<1, blk, 0, stream>>>(part, 256, sc + 9);        // w_loss
  rowreduce_kernel<false><<<N1C, blk, 0, stream>>>(SH, nullptr, afv, N2C, 0.0f);
  colreduce_kernel<false><<<g1(N2C), blk, 0, stream>>>(SH, nullptr, bfv, N1C, N2C, 0.0f);
  rowreduce_kernel<true ><<<N1C, blk, 0, stream>>>(C1, afv, u2v, N1C, 0.0f);
  colreduce_kernel<true ><<<g1(N2C), blk, 0, stream>>>(C2, bfv, v2v, N2C, N2C, 0.0f);
  gemm_wmma_f32<POST_NONE, false><<<gg(N1C, N2C), blk, 0, stream>>>(
      C1, SH, T1, N1C, N1C, N2C, N2C, nullptr, nullptr, nullptr, nullptr, nullptr);
  gemm_wmma_f32<POST_NONE, true><<<gg(N1C, N2C), blk, 0, stream>>>(
      T1, C2, Q, N2C, N2C, N2C, N2C, nullptr, nullptr, nullptr, nullptr, nullptr);
  gw_stage1<<<256, blk, 0, stream>>>(Q, SH, u2v, v2v, part, N2C, NE);
  reduce_final_kernel<<<1, blk, 0, stream>>>(part, 256, sc + 10);       // gw_loss

  loss_finalize_kernel<<<1, 32, 0, stream>>>(sc, out, NE);
  copy_kernel<<<g1(NE), blk, 0, stream>>>(out + 1, S, NE);              // outputs: [loss, s..., lam]
}